// GaussianOverlapLoss_7438883357516
// MI455X (gfx1250) — compile-verified
//
#include <hip/hip_runtime.h>
#include <math.h>

// ---------------- constants from the reference ----------------
#define N_PTS 8192
#define C_ALPHA 0.01f
#define C_BETA  0.6065f
#define C_LAMB  0.0001f
#define C_EPS   1e-07f

#define TI 256   // i-rows per block
#define TJ 256   // j-cols per block (one LDS tile)

// -0.125 * log2(e): fold exp(-0.125*q) into a single v_exp_f32 (exp2)
#define NEG_EIGHTH_LOG2E (-0.18033688011112042f)

typedef __attribute__((ext_vector_type(2))) float v2f;
typedef __attribute__((ext_vector_type(8))) float v8f;

// ---------------------------------------------------------------
// Kernel 1: per-Gaussian eigenvalue/log1p term  llamb[i]
// (O(N), cold path: keep fully precise libm)
// ---------------------------------------------------------------
__global__ __launch_bounds__(256) void k_pre(const float* __restrict__ sigma,
                                             float* __restrict__ llamb) {
  int i = blockIdx.x * blockDim.x + threadIdx.x;
  if (i >= N_PTS) return;
  float a = sigma[i * 4 + 0];
  float c = sigma[i * 4 + 2];
  float d = sigma[i * 4 + 3];
  float tr   = a + d;
  float disc = sqrtf(fmaxf((a - d) * (a - d) + 4.0f * c * c, 0.0f));
  float e0 = (tr - disc) * 0.5f;
  float e1 = (tr + disc) * 0.5f;
  float L0 = sqrtf(fmaxf(e0, C_EPS));
  float L1 = sqrtf(fmaxf(e1, C_EPS));
  llamb[i] = log1pf(fabsf(L0)) + log1pf(fabsf(L1));
}

// ---------------------------------------------------------------
// Kernel 2: 256x256 pair tiles. One i per thread, j staged in LDS
// with covariances pre-halved so the matrix average is a plain add.
// Hot loop uses 1-ulp HW transcendentals (v_rcp/v_sqrt/v_exp).
// Wave reduction via V_WMMA_F32_16X16X4_F32 (B = ones).
// ---------------------------------------------------------------
__global__ __launch_bounds__(256) void k_pairs(const float* __restrict__ mu,
                                               const float* __restrict__ sigma,
                                               float* __restrict__ partial) {
  __shared__ float4 sj[TJ * 2];   // {x, y, a/2, b/2} , {c/2, d/2, det, pad}
  __shared__ float  wsum[8];

  const int tid = threadIdx.x;
  const int gi  = blockIdx.x * TI + tid;
  const int j0  = blockIdx.y * TJ;

  // own row-i data (covariance pre-halved; det kept full)
  const float xi = mu[gi * 2 + 0];
  const float yi = mu[gi * 2 + 1];
  const float ai = sigma[gi * 4 + 0];
  const float bi = sigma[gi * 4 + 1];
  const float ci = sigma[gi * 4 + 2];
  const float di = sigma[gi * 4 + 3];
  const float deti = ai * di - bi * ci;
  const float hai = 0.5f * ai, hbi = 0.5f * bi, hci = 0.5f * ci, hdi = 0.5f * di;

  // cooperative stage of the j tile (256 entries, 8 KB LDS)
  {
    const int j  = j0 + tid;
    float xj = mu[j * 2 + 0];
    float yj = mu[j * 2 + 1];
    float aj = sigma[j * 4 + 0];
    float bj = sigma[j * 4 + 1];
    float cj = sigma[j * 4 + 2];
    float dj = sigma[j * 4 + 3];
    float detj = aj * dj - bj * cj;
    sj[2 * tid + 0] = make_float4(xj, yj, 0.5f * aj, 0.5f * bj);
    sj[2 * tid + 1] = make_float4(0.5f * cj, 0.5f * dj, detj, 0.0f);
  }
  __syncthreads();

  float acc = 0.0f;
#pragma unroll 4
  for (int jj = 0; jj < TJ; ++jj) {
    // uniform LDS address across the wave -> broadcast ds_load_b128
    float4 p0 = sj[2 * jj + 0];
    float4 p1 = sj[2 * jj + 1];

    float aM = hai + p0.z;                          // (ai + aj) * 0.5
    float bM = hbi + p0.w;
    float cM = hci + p1.x;
    float dM = hdi + p1.y;
    float detM  = aM * dM - bM * cM;                // >= 0.25 (AA^T + 0.5 I)
    float rdetM = __builtin_amdgcn_rcpf(detM);      // v_rcp_f32, <=1 ulp

    float dx = p0.x - xi;
    float dy = p0.y - yi;
    float quad = (dM * dx * dx - (bM + cM) * dx * dy + aM * dy * dy) * rdetM;
    float t0   = __builtin_amdgcn_exp2f(quad * NEG_EIGHTH_LOG2E);       // v_exp_f32
    float t1   = __builtin_amdgcn_sqrtf(fmaxf(deti * p1.z, C_EPS));     // v_sqrt_f32
    float coef = t0 * __builtin_amdgcn_sqrtf(fmaxf(t1 * rdetM, C_EPS)); // v_sqrt_f32
    if (gi == j0 + jj) coef = 0.0f;                 // (1 - eye) mask

    float x = coef - C_BETA;
    acc += fmaxf(x, C_ALPHA * x);                   // leaky_relu; +BETA*ALPHA hoisted
  }
  // hoisted constant: TJ pairs each contribute +BETA*ALPHA
  acc += (float)TJ * (C_BETA * C_ALPHA);

  // ---- wave32 reduction with one f32 WMMA: D = A(16x4) * ones(4x16) ----
  // A: VGPR0 holds acc (K=0 lanes 0-15, K=2 lanes 16-31), VGPR1 = 0.
  // D[m][n] = acc_m + acc_{m+16}; per-lane sum of the 8 D regs covers half
  // the M rows; pairing lane l with lane l^16 yields the exact wave sum.
  v2f Aop = {acc, 0.0f};
  v2f Bop = {1.0f, 1.0f};
  v8f Cop = {};
  v8f Dop = __builtin_amdgcn_wmma_f32_16x16x4_f32(
      /*neg_a=*/false, Aop, /*neg_b=*/false, Bop,
      /*c_mod=*/(short)0, Cop, /*reuse_a=*/false, /*reuse_b=*/false);

  float s = Dop[0] + Dop[1] + Dop[2] + Dop[3] + Dop[4] + Dop[5] + Dop[6] + Dop[7];
  float waveTotal = s + __shfl_xor(s, 16, 32);

  const int lane = tid & 31;
  const int wave = tid >> 5;
  if (lane == 0) wsum[wave] = waveTotal;
  __syncthreads();

  if (tid == 0) {
    float p = 0.0f;
    for (int w = 0; w < 8; ++w) p += wsum[w];   // fixed order: deterministic
    partial[blockIdx.y * gridDim.x + blockIdx.x] = p;
  }
}

// ---------------------------------------------------------------
// Kernel 3: deterministic final reduction (single block, doubles)
// ---------------------------------------------------------------
__global__ __launch_bounds__(256) void k_final(const float* __restrict__ llamb,
                                               const float* __restrict__ partial,
                                               int nPartial,
                                               float* __restrict__ out) {
  __shared__ double sd[256];
  const int tid = threadIdx.x;

  double s_l = 0.0;
  for (int i = tid; i < N_PTS; i += 256) s_l += (double)llamb[i];
  double s_p = 0.0;
  for (int i = tid; i < nPartial; i += 256) s_p += (double)partial[i];

  // loss = LAMB * mean(llamb over 2N entries) + pair_sum / N   (LOSS_WEIGHT = 1)
  sd[tid] = s_l * ((double)C_LAMB / (2.0 * (double)N_PTS)) + s_p / (double)N_PTS;
  __syncthreads();

  for (int st = 128; st > 0; st >>= 1) {
    if (tid < st) sd[tid] += sd[tid + st];
    __syncthreads();
  }
  if (tid == 0) out[0] = (float)sd[0];
}

// ---------------------------------------------------------------
extern "C" void kernel_launch(void* const* d_in, const int* in_sizes, int n_in,
                              void* d_out, int out_size, void* d_ws, size_t ws_size,
                              hipStream_t stream) {
  const float* mu    = (const float*)d_in[0];   // (N,2) float32
  const float* sigma = (const float*)d_in[1];   // (N,2,2) float32
  float* out = (float*)d_out;                   // scalar float32

  // workspace layout: llamb[N] | partial[1024]   (~36 KB)
  float* llamb   = (float*)d_ws;
  float* partial = llamb + N_PTS;
  const int nPartial = (N_PTS / TI) * (N_PTS / TJ);  // 1024

  k_pre<<<N_PTS / 256, 256, 0, stream>>>(sigma, llamb);

  dim3 grid(N_PTS / TI, N_PTS / TJ);  // 32 x 32 = 1024 blocks
  k_pairs<<<grid, 256, 0, stream>>>(mu, sigma, partial);

  k_final<<<1, 256, 0, stream>>>(llamb, partial, nPartial, out);
}